// BinCat_62551903699054
// MI455X (gfx1250) — compile-verified
//
#include <hip/hip_runtime.h>
#include <hip/hip_bf16.h>

// BinCat gather: idx = ~packed_bits(x) & 0xFFFFF; out[row] = cats[idx]
// Pure HBM-bound gather (~37 MB @ 23.3 TB/s ~ 1.6 us). No matmul content ->
// no WMMA; instead exercise CDNA5 async global<->LDS DMA (ASYNCcnt) path:
//   global_load_async_to_lds_b128  (per-lane gather, 32x16B per instr)
//   s_wait_asynccnt 0
//   global_store_async_from_lds_b128 (coalesced writeback, no VGPR staging)

#define LENGTH          20
#define DIM             64
#define ROW_BYTES       (DIM * 4)              // 256 B per embedding row
#define ROWS_PER_WAVE   32                     // one idx per lane
#define WAVES_PER_BLOCK 4
#define THREADS         (WAVES_PER_BLOCK * 32) // 128, wave32
#define ROWS_PER_BLOCK  (ROWS_PER_WAVE * WAVES_PER_BLOCK) // 128
#define LDS_PER_WAVE    (ROWS_PER_WAVE * ROW_BYTES)       // 8 KB

__global__ __launch_bounds__(THREADS)
void bincat_async_gather(const int* __restrict__ x,
                         const float* __restrict__ cats,
                         float* __restrict__ out)
{
    __shared__ __align__(16) char sh[WAVES_PER_BLOCK * LDS_PER_WAVE]; // 32 KB

    const unsigned tid  = threadIdx.x;
    const unsigned lane = tid & 31u;
    const unsigned wave = tid >> 5;
    const unsigned base = blockIdx.x * ROWS_PER_BLOCK + wave * ROWS_PER_WAVE;
    const unsigned row  = base + lane;          // this lane's row

    // ---- Phase A: pack 20 bits, complement -> embedding index ----------
    // x[row] is 80 B, 16 B aligned (80 % 16 == 0): 5 x b128 loads.
    const int4* px = reinterpret_cast<const int4*>(x + (size_t)row * LENGTH);
    int4 a = px[0], b = px[1], c = px[2], d = px[3], e = px[4];
    unsigned bits =
        ((unsigned)a.x << 19) | ((unsigned)a.y << 18) | ((unsigned)a.z << 17) | ((unsigned)a.w << 16) |
        ((unsigned)b.x << 15) | ((unsigned)b.y << 14) | ((unsigned)b.z << 13) | ((unsigned)b.w << 12) |
        ((unsigned)c.x << 11) | ((unsigned)c.y << 10) | ((unsigned)c.z <<  9) | ((unsigned)c.w <<  8) |
        ((unsigned)d.x <<  7) | ((unsigned)d.y <<  6) | ((unsigned)d.z <<  5) | ((unsigned)d.w <<  4) |
        ((unsigned)e.x <<  3) | ((unsigned)e.y <<  2) | ((unsigned)e.z <<  1) | ((unsigned)e.w      );
    int idx = (int)((~bits) & 0xFFFFFu);        // sum_j (1-x_j)*2^(19-j)

    // Wave-relative LDS byte offset: low 32 bits of generic LDS pointer.
    const unsigned ldsWaveBase =
        (unsigned)(uintptr_t)(&sh[0]) + wave * LDS_PER_WAVE;
    const unsigned half = lane >> 4;            // 0 | 1 : which row of the pair
    const unsigned quad = lane & 15u;           // 16 B chunk within the row

    // ---- Phase B: async gather cats rows -> LDS ------------------------
    // Step k: lanes 0-15 fetch row (2k), lanes 16-31 fetch row (2k+1);
    // each 16-lane group reads one contiguous 256 B row (2 cache lines).
#pragma unroll
    for (int k = 0; k < 16; ++k) {
        int ridx = __shfl(idx, 2 * k + (int)half, 32);         // wave32 bcast
        int goff = ridx * ROW_BYTES + (int)(quad * 16u);       // < 2^28
        unsigned laddr = ldsWaveBase + (unsigned)k * 512u + lane * 16u;
        asm volatile("global_load_async_to_lds_b128 %0, %1, %2"
                     :: "v"(laddr), "v"(goff), "s"(cats) : "memory");
    }
    asm volatile("s_wait_asynccnt 0" ::: "memory");            // gather done

    // ---- Phase C: async store LDS -> out (512 B contiguous / instr) ----
#pragma unroll
    for (int k = 0; k < 16; ++k) {
        unsigned laddr = ldsWaveBase + (unsigned)k * 512u + lane * 16u;
        int ooff = (int)(base * ROW_BYTES + (unsigned)k * 512u + lane * 16u);
        asm volatile("global_store_async_from_lds_b128 %0, %1, %2"
                     :: "v"(ooff), "v"(laddr), "s"(out) : "memory");
    }
    asm volatile("s_wait_asynccnt 0" ::: "memory");  // (endpgm also waits idle)
}

extern "C" void kernel_launch(void* const* d_in, const int* in_sizes, int n_in,
                              void* d_out, int out_size, void* d_ws, size_t ws_size,
                              hipStream_t stream)
{
    (void)n_in; (void)out_size; (void)d_ws; (void)ws_size;
    const int*   x    = (const int*)  d_in[0];   // (4096, 16, 20) int32 {0,1}
    const float* cats = (const float*)d_in[1];   // (2^20, 64) fp32
    float*       out  = (float*)      d_out;     // (4096, 16, 64) fp32

    const int rows = in_sizes[0] / LENGTH;       // 65536 (multiple of 128)
    const int grid = rows / ROWS_PER_BLOCK;      // 512 blocks
    bincat_async_gather<<<grid, THREADS, 0, stream>>>(x, cats, out);
}